// ConvIntrinsic_47270410060382
// MI455X (gfx1250) — compile-verified
//
#include <hip/hip_runtime.h>

// ConvIntrinsic on MI455X (gfx1250): restructured as
//   sig (6000x5120) = gather+barycentric            [kernel 1, mem-bound]
//   Wc  (5120x2048) = interp (x) rotated kernel_wts [kernel 2, small]
//   out = epilogue( sig @ Wc )                      [kernel 3, f32 WMMA GEMM]
// GEMM: M=6000, K=5120, N=2048 (cols = rot*256 + k*128 + o); fused
// relu(s_k0+40b0)+relu(s_k1+40b1) epilogue in registers.

typedef float v2f __attribute__((ext_vector_type(2)));
typedef float v8f __attribute__((ext_vector_type(8)));

#define NV    6000
#define FF    128
#define QQ    40              // R*A
#define KDIM  (QQ * FF)       // 5120
#define NROT  8
#define KK    2
#define OO    128
#define NCOL  (NROT * KK * OO) // 2048
#define BM    256
#define KC    32

// ---------------- Kernel 1: gather + barycentric interpolation ----------------
// sig[(n*40+q)*128 + f] = sum_j w[n,q,j] * mesh_signal[idx[n,q,j], f]
__global__ __launch_bounds__(256) void k_gather(const float* __restrict__ ms,
                                                const float* __restrict__ bary,
                                                float* __restrict__ sig, int total) {
  int t = blockIdx.x * 256 + threadIdx.x;
  if (t >= total) return;
  int f  = t & (FF - 1);
  int nq = t >> 7;                       // n*40 + q
  const float* b = bary + (size_t)nq * 6; // (idx,w) x 3
  float acc = 0.f;
#pragma unroll
  for (int j = 0; j < 3; ++j) {
    int   idx = (int)b[2 * j];
    float w   = b[2 * j + 1];
    acc = fmaf(w, ms[(size_t)idx * FF + f], acc);
  }
  sig[t] = acc;
}

// ---------------- Kernel 2: build combined weight matrix Wc ----------------
// Wc[(q*128+f) * 2048 + rot*256 + k*128 + o]
//   = sum_{r,a} interp[r,a,q] * kw[r, (a+rot)&7, k, o, f]
// grid.x = 1024 : (rot, k, o-block of 16, f-block of 16)
__global__ __launch_bounds__(256) void k_wc(const float* __restrict__ kw,
                                            const float* __restrict__ interp,
                                            float* __restrict__ Wc) {
  __shared__ float s_interp[QQ * QQ];    // [ra][q], 1600 floats
  __shared__ float s_kw[QQ][16][17];     // [ra][oo][ff], padded stride 17

  int b     = blockIdx.x;
  int fblk  = b & 7;
  int oblk  = (b >> 3) & 7;
  int k     = (b >> 6) & 1;
  int rot   = b >> 7;
  int fbase = fblk * 16;
  int obase = oblk * 16;
  int cbase = rot * 256 + k * 128 + obase;

  int tid = threadIdx.x;
  for (int i = tid; i < QQ * QQ; i += 256) s_interp[i] = interp[i];

  // load kw tile: ff fastest across lanes -> coalesced
  {
    int lff = tid & 15;
    int loo = tid >> 4;
#pragma unroll 4
    for (int ra = 0; ra < QQ; ++ra) {
      int r  = ra >> 3, a = ra & 7;
      int a2 = (a + rot) & 7;
      size_t off = ((size_t)(((r * 8 + a2) * 2 + k) * 128 + obase + loo)) * 128 +
                   fbase + lff;
      s_kw[ra][loo][lff] = kw[off];
    }
  }
  __syncthreads();

  // compute: oo fastest across lanes -> coalesced global writes
  int oo = tid & 15;
  int ff = tid >> 4;
  for (int q = 0; q < QQ; ++q) {
    float acc = 0.f;
#pragma unroll
    for (int ra = 0; ra < QQ; ++ra)
      acc = fmaf(s_interp[ra * QQ + q], s_kw[ra][oo][ff], acc);
    int p = q * FF + fbase + ff;
    Wc[(size_t)p * NCOL + cbase + oo] = acc;
  }
}

// ---------------- Kernel 3: f32 WMMA GEMM + fused epilogue ----------------
// grid.x = 64 (rot*8 + o_tile), grid.y = ceil(6000/256). 8 waves/WG.
// Wave w: rows [w*32, w*32+32) of the 256-row block; 16 o-cols; both k slices.
__global__ __launch_bounds__(256) void k_gemm(const float* __restrict__ A,
                                              const float* __restrict__ B,
                                              const float* __restrict__ bias,
                                              float* __restrict__ out) {
  __shared__ float sA[BM][KC + 4];       // stride 36 floats: conflict-free, 16B rows
  __shared__ float sB0[KC][16];          // k=0 slice
  __shared__ float sB1[KC][16];          // k=1 slice

  const int tid  = threadIdx.x;
  const int wave = tid >> 5;
  const int lane = tid & 31;
  const int mrow = lane & 15;            // M (for A) / N (for B,C) index
  const int half = lane >> 4;            // K half-select per ISA 16x4 layout

  const int otile = blockIdx.x & 7;
  const int rot   = blockIdx.x >> 3;
  const int row0  = blockIdx.y * BM;
  const int obase = otile * 16;

  const float* B0 = B + rot * 256 + obase;   // cols (rot, k=0, obase..+15)
  const float* B1 = B0 + 128;                // cols (rot, k=1, obase..+15)

  v8f acc00 = {}, acc01 = {}, acc10 = {}, acc11 = {};

  const int arow0 = wave * 32 + mrow;
  const int arow1 = arow0 + 16;

  for (int kc = 0; kc < KDIM; kc += KC) {
    // stage A: 256x32 f32 (2048 float4, 8 per thread)
#pragma unroll
    for (int i = 0; i < 8; ++i) {
      int l  = tid + i * 256;
      int r  = l >> 3;
      int c4 = (l & 7) << 2;
      float4 v = make_float4(0.f, 0.f, 0.f, 0.f);
      int gr = row0 + r;
      if (gr < NV) v = *(const float4*)(A + (size_t)gr * KDIM + kc + c4);
      *(float4*)&sA[r][c4] = v;
    }
    // stage B: two 32x16 tiles (128 float4 each)
    {
      int tsel = tid >> 7;
      int idx  = tid & 127;
      int r    = idx >> 2;
      int c4   = (idx & 3) << 2;
      const float* src = (tsel ? B1 : B0) + (size_t)(kc + r) * NCOL + c4;
      float4 v = *(const float4*)src;
      float* dst = tsel ? &sB1[r][c4] : &sB0[r][c4];
      *(float4*)dst = v;
    }
    __syncthreads();

#pragma unroll
    for (int kk = 0; kk < KC; kk += 4) {
      int klo = kk + (half << 1);        // lanes 0-15: K{0,1}; lanes 16-31: K{2,3}
      v2f a0, a1, b0, b1;
      a0.x = sA[arow0][klo]; a0.y = sA[arow0][klo + 1];
      a1.x = sA[arow1][klo]; a1.y = sA[arow1][klo + 1];
      b0.x = sB0[klo][mrow]; b0.y = sB0[klo + 1][mrow];
      b1.x = sB1[klo][mrow]; b1.y = sB1[klo + 1][mrow];
      acc00 = __builtin_amdgcn_wmma_f32_16x16x4_f32(false, a0, false, b0,
                                                    (short)0, acc00, false, false);
      acc01 = __builtin_amdgcn_wmma_f32_16x16x4_f32(false, a0, false, b1,
                                                    (short)0, acc01, false, false);
      acc10 = __builtin_amdgcn_wmma_f32_16x16x4_f32(false, a1, false, b0,
                                                    (short)0, acc10, false, false);
      acc11 = __builtin_amdgcn_wmma_f32_16x16x4_f32(false, a1, false, b1,
                                                    (short)0, acc11, false, false);
    }
    __syncthreads();
  }

  // fused epilogue: relu(s_k0 + 40*b0) + relu(s_k1 + 40*b1), write (n, rot, o)
  const int   o   = obase + mrow;
  const float bv0 = 40.0f * bias[o];
  const float bv1 = 40.0f * bias[OO + o];
#pragma unroll
  for (int i = 0; i < 8; ++i) {
    int m0 = row0 + wave * 32 + i + (half << 3);  // C layout: VGPR i -> M=i / i+8
    int m1 = m0 + 16;
    float r0 = fmaxf(acc00[i] + bv0, 0.f) + fmaxf(acc01[i] + bv1, 0.f);
    float r1 = fmaxf(acc10[i] + bv0, 0.f) + fmaxf(acc11[i] + bv1, 0.f);
    if (m0 < NV) out[(size_t)m0 * (NROT * OO) + rot * OO + o] = r0;
    if (m1 < NV) out[(size_t)m1 * (NROT * OO) + rot * OO + o] = r1;
  }
}

extern "C" void kernel_launch(void* const* d_in, const int* in_sizes, int n_in,
                              void* d_out, int out_size, void* d_ws, size_t ws_size,
                              hipStream_t stream) {
  const float* ms     = (const float*)d_in[0];  // (6000,128)
  const float* bary   = (const float*)d_in[1];  // (6000,5,8,3,2)
  const float* kw     = (const float*)d_in[2];  // (5,8,2,128,128)
  const float* bias   = (const float*)d_in[3];  // (2,128)
  const float* interp = (const float*)d_in[4];  // (5,8,40)
  float* out = (float*)d_out;                   // (6000,8,128)

  float* sig = (float*)d_ws;                                        // 122.88 MB
  float* Wc  = (float*)((char*)d_ws + (size_t)NV * KDIM * sizeof(float)); // 41.94 MB

  int total = NV * QQ * FF;
  k_gather<<<(total + 255) / 256, 256, 0, stream>>>(ms, bary, sig, total);
  k_wc<<<NROT * KK * 8 * 8, 256, 0, stream>>>(kw, interp, Wc);
  dim3 grid(64, (NV + BM - 1) / BM);
  k_gemm<<<grid, 256, 0, stream>>>(sig, Wc, bias, out);
}